// BoundarySuppressionWithSmoothing_14620068675670
// MI455X (gfx1250) — compile-verified
//
#include <hip/hip_runtime.h>
#include <hip/hip_bf16.h>
#include <stdint.h>

typedef __attribute__((ext_vector_type(2))) float v2f;
typedef __attribute__((ext_vector_type(8))) float v8f;

#define BATCH 4
#define HH 1024
#define WW 2048
#define PLANE ((size_t)HH * WW)

#define TDIM 16
#define KPAD 18
#define TW (TDIM + 2 * KPAD) /* 52 */

__device__ __forceinline__ int clampi(int v, int lo, int hi) {
    return v < lo ? lo : (v > hi ? hi : v);
}

// ---------------------------------------------------------------------------
// Kernel 1: boundaries -> initial level map (0 = boundary, 4 = far)
// boundary = (max over cross neighborhood != min over 3x3), edge-clamped.
// ---------------------------------------------------------------------------
__global__ void k_init_level(const int* __restrict__ pred,
                             unsigned char* __restrict__ lvl) {
    int x = blockIdx.x * blockDim.x + threadIdx.x;
    int y = blockIdx.y;
    int b = blockIdx.z;
    size_t base = (size_t)b * PLANE;
    const int* P = pred + base;

    int xm = x > 0 ? x - 1 : 0, xp = x < WW - 1 ? x + 1 : x;
    int ym = y > 0 ? y - 1 : 0, yp = y < HH - 1 ? y + 1 : y;

    int c  = P[(size_t)y * WW + x];
    int up = P[(size_t)ym * WW + x], dn = P[(size_t)yp * WW + x];
    int lf = P[(size_t)y * WW + xm], rt = P[(size_t)y * WW + xp];
    int ul = P[(size_t)ym * WW + xm], ur = P[(size_t)ym * WW + xp];
    int dl = P[(size_t)yp * WW + xm], dr = P[(size_t)yp * WW + xp];

    int cross = max(c, max(max(up, dn), max(lf, rt)));
    int mn = min(min(min(c, up), min(dn, lf)),
                 min(min(rt, ul), min(ur, dl)));
    mn = min(mn, dr);

    lvl[base + (size_t)y * WW + x] = (cross != mn) ? 0 : 4;
}

// ---------------------------------------------------------------------------
// Kernel 2: one Chebyshev-distance relaxation pass:
//           lvl' = min(lvl, 1 + min over clamped 3x3 of lvl)
// Three passes give lvl = min(chebyshev_dist_to_boundary, 4).
// ---------------------------------------------------------------------------
__global__ void k_relax(const unsigned char* __restrict__ in,
                        unsigned char* __restrict__ out) {
    int x = blockIdx.x * blockDim.x + threadIdx.x;
    int y = blockIdx.y;
    int b = blockIdx.z;
    size_t base = (size_t)b * PLANE;

    int xm = x > 0 ? x - 1 : 0, xp = x < WW - 1 ? x + 1 : x;
    int ym = y > 0 ? y - 1 : 0, yp = y < HH - 1 ? y + 1 : y;

    int m = in[base + (size_t)ym * WW + xm];
    m = min(m, (int)in[base + (size_t)ym * WW + x]);
    m = min(m, (int)in[base + (size_t)ym * WW + xp]);
    m = min(m, (int)in[base + (size_t)y  * WW + xm]);
    int c = in[base + (size_t)y * WW + x];
    m = min(m, c);
    m = min(m, (int)in[base + (size_t)y  * WW + xp]);
    m = min(m, (int)in[base + (size_t)yp * WW + xm]);
    m = min(m, (int)in[base + (size_t)yp * WW + x]);
    m = min(m, (int)in[base + (size_t)yp * WW + xp]);

    out[base + (size_t)y * WW + x] = (unsigned char)min(c, m + 1);
}

// ---------------------------------------------------------------------------
// Kernel 3: one masked 3x3 box-smoothing iteration.
// mask = (lvl > thresh). mask==1 -> keep. mask==0 -> average of masked
// neighbors (clamped, duplicates counted), or keep if no masked neighbor.
// ---------------------------------------------------------------------------
__global__ void k_smooth(const float* __restrict__ in,
                         const unsigned char* __restrict__ lvl,
                         float* __restrict__ out, int thresh) {
    int x = blockIdx.x * blockDim.x + threadIdx.x;
    int y = blockIdx.y;
    int b = blockIdx.z;
    size_t base = (size_t)b * PLANE;
    size_t pc = base + (size_t)y * WW + x;

    float vin = in[pc];
    float res = vin;
    if (lvl[pc] <= thresh) {
        int xs[3], ys[3];
        xs[0] = x > 0 ? x - 1 : 0; xs[1] = x; xs[2] = x < WW - 1 ? x + 1 : x;
        ys[0] = y > 0 ? y - 1 : 0; ys[1] = y; ys[2] = y < HH - 1 ? y + 1 : y;
        float s = 0.0f;
        int cnt = 0;
#pragma unroll
        for (int i = 0; i < 3; i++) {
#pragma unroll
            for (int j = 0; j < 3; j++) {
                size_t q = base + (size_t)ys[i] * WW + xs[j];
                int m = (lvl[q] > thresh) ? 1 : 0;
                cnt += m;
                s += m ? in[q] : 0.0f;
            }
        }
        if (cnt != 0) res = s / (float)cnt;
    }
    out[pc] = res;
}

// ---------------------------------------------------------------------------
// Gaussian, horizontal pass as banded GEMM on WMMA.
// One wave (32 threads) computes a 16x16 tile:
//   D(16x16) = A(16x52) x Band(52x16),  A[r,m] = in[r, clamp(C0-18+m)]
//   Band[m,c] = w1d[j] iff m == c + 6j  (built by scatter, no per-step ALU)
// w1d = column sums of the separable 7x7 Gaussian.
// Inner loop is pure ds_load + v_wmma; EXEC is all-ones throughout it.
// ---------------------------------------------------------------------------
__global__ void __launch_bounds__(32)
k_gauss_h(const float* __restrict__ in, const float* __restrict__ gk,
          float* __restrict__ out) {
    __shared__ float tile[TDIM * TW];   // [16 rows][52 cols]
    __shared__ float band[TW * TDIM];   // [52 rows][16 cols]
    int lane = threadIdx.x;
    int C0 = blockIdx.x * TDIM;
    int R0 = blockIdx.y * TDIM;
    size_t base = (size_t)blockIdx.z * PLANE;

    // 1D horizontal weights = column sums (uniform scalar loads).
    float w[7];
#pragma unroll
    for (int j = 0; j < 7; j++) {
        float s = 0.0f;
#pragma unroll
        for (int i = 0; i < 7; i++) s += gk[i * 7 + j];
        w[j] = s;
    }

    // Zero band, then scatter the 7 taps per column.
    for (int e = lane; e < TW * TDIM; e += 32) band[e] = 0.0f;
    // Input tile with clamped columns (no div/mod: explicit row loop).
#pragma unroll
    for (int r = 0; r < TDIM; r++) {
        const float* row = in + base + (size_t)(R0 + r) * WW;
        for (int m = lane; m < TW; m += 32)
            tile[r * TW + m] = row[clampi(C0 - KPAD + m, 0, WW - 1)];
    }
    __syncthreads();
    if (lane < TDIM) {
#pragma unroll
        for (int j = 0; j < 7; j++)
            band[(lane + 6 * j) * TDIM + lane] = w[j];
    }
    __syncthreads();

    int lg = lane >> 4;   // lane-group: 0 or 1
    int lc = lane & 15;
    v8f acc = {};
#pragma unroll
    for (int kk = 0; kk < 13; kk++) {
        int kb = kk * 4 + 2 * lg;     // this lane's K base (doc'd A layout)
        v2f a, bm;
        a.x = tile[lc * TW + kb];
        a.y = tile[lc * TW + kb + 1];
        bm.x = band[kb * TDIM + lc];
        bm.y = band[(kb + 1) * TDIM + lc];
        acc = __builtin_amdgcn_wmma_f32_16x16x4_f32(
            false, a, false, bm, (short)0, acc, false, false);
    }
#pragma unroll
    for (int v = 0; v < 8; v++) {
        int r = v + 8 * lg;           // doc'd D layout
        out[base + (size_t)(R0 + r) * WW + (C0 + lc)] = acc[v];
    }
}

// ---------------------------------------------------------------------------
// Gaussian, vertical pass:  D(16x16) = Band(16x52) x Bt(52x16),
//   Bt[m,c] = in[clamp(R0-18+m), C0+c],  Band[r, r+6j] = w1d[j]
// w1d = row sums.
// ---------------------------------------------------------------------------
__global__ void __launch_bounds__(32)
k_gauss_v(const float* __restrict__ in, const float* __restrict__ gk,
          float* __restrict__ out) {
    __shared__ float tile[TW * TDIM];   // [52 rows][16 cols]
    __shared__ float band[TDIM * TW];   // [16 rows][52 cols]
    int lane = threadIdx.x;
    int C0 = blockIdx.x * TDIM;
    int R0 = blockIdx.y * TDIM;
    size_t base = (size_t)blockIdx.z * PLANE;

    // 1D vertical weights = row sums.
    float w[7];
#pragma unroll
    for (int i = 0; i < 7; i++) {
        float s = 0.0f;
#pragma unroll
        for (int j = 0; j < 7; j++) s += gk[i * 7 + j];
        w[i] = s;
    }

    for (int e = lane; e < TDIM * TW; e += 32) band[e] = 0.0f;
    // Input tile with clamped rows; 2 rows per pass, coalesced 64B per row.
    {
        int lg = lane >> 4, lc = lane & 15;
#pragma unroll
        for (int m0 = 0; m0 < TW; m0 += 2) {
            int m = m0 + lg;
            int gr = clampi(R0 - KPAD + m, 0, HH - 1);
            tile[m * TDIM + lc] = in[base + (size_t)gr * WW + (C0 + lc)];
        }
    }
    __syncthreads();
    if (lane < TDIM) {
#pragma unroll
        for (int j = 0; j < 7; j++)
            band[lane * TW + (lane + 6 * j)] = w[j];
    }
    __syncthreads();

    int lg = lane >> 4;
    int lc = lane & 15;
    v8f acc = {};
#pragma unroll
    for (int kk = 0; kk < 13; kk++) {
        int kb = kk * 4 + 2 * lg;
        v2f a, bm;
        a.x = band[lc * TW + kb];
        a.y = band[lc * TW + kb + 1];
        bm.x = tile[kb * TDIM + lc];
        bm.y = tile[(kb + 1) * TDIM + lc];
        acc = __builtin_amdgcn_wmma_f32_16x16x4_f32(
            false, a, false, bm, (short)0, acc, false, false);
    }
#pragma unroll
    for (int v = 0; v < 8; v++) {
        int r = v + 8 * lg;
        out[base + (size_t)(R0 + r) * WW + (C0 + lc)] = acc[v];
    }
}

// ---------------------------------------------------------------------------
// Host-side launch chain.
// ws layout: [ float bufA : N floats ][ uint8 lvlA : N ][ uint8 lvlB : N ]
// d_out serves as the float pong buffer; chain ends writing d_out.
// ---------------------------------------------------------------------------
extern "C" void kernel_launch(void* const* d_in, const int* in_sizes, int n_in,
                              void* d_out, int out_size, void* d_ws, size_t ws_size,
                              hipStream_t stream) {
    const float* x    = (const float*)d_in[0];
    const int*   pred = (const int*)d_in[1];
    const float* gk   = (const float*)d_in[3];   // 7x7 gaussian (separable)

    const size_t N = (size_t)BATCH * PLANE;
    float* bufA = (float*)d_ws;
    unsigned char* lvlA = (unsigned char*)d_ws + N * sizeof(float);
    unsigned char* lvlB = lvlA + N;
    float* bufO = (float*)d_out;

    dim3 blk(256, 1, 1);
    dim3 grd(WW / 256, HH, BATCH);

    // Level map: boundaries then 3 relaxation passes -> final in lvlB.
    k_init_level<<<grd, blk, 0, stream>>>(pred, lvlA);
    k_relax<<<grd, blk, 0, stream>>>(lvlA, lvlB);
    k_relax<<<grd, blk, 0, stream>>>(lvlB, lvlA);
    k_relax<<<grd, blk, 0, stream>>>(lvlA, lvlB);

    // Four masked box-smoothing iterations (thresh = 3 - it).
    k_smooth<<<grd, blk, 0, stream>>>(x,    lvlB, bufA, 3);
    k_smooth<<<grd, blk, 0, stream>>>(bufA, lvlB, bufO, 2);
    k_smooth<<<grd, blk, 0, stream>>>(bufO, lvlB, bufA, 1);
    k_smooth<<<grd, blk, 0, stream>>>(bufA, lvlB, bufO, 0);

    // Separable dilated 7x7 Gaussian on the WMMA path.
    dim3 gblk(32, 1, 1);
    dim3 ggrd(WW / TDIM, HH / TDIM, BATCH);
    k_gauss_h<<<ggrd, gblk, 0, stream>>>(bufO, gk, bufA);
    k_gauss_v<<<ggrd, gblk, 0, stream>>>(bufA, gk, bufO);
}